// Loss_34230889349355
// MI455X (gfx1250) — compile-verified
//
#include <hip/hip_runtime.h>
#include <math.h>

#define B_ 8
#define N_ 4096
#define P_ 16
#define S_ 200
#define SPAD 208          // 13 * 16
#define STILES 13
#define JTILES 16         // 16 n-tiles of 16 points per wave => 256 points/wave
#define NCHUNK 16         // chunks per (b,p): 16 * 256 = 4096
#define NWAVES (B_ * P_ * NCHUNK)   // 2048
#define EXIST_THRED_ 24.0f
#define BIGD 1e30f

typedef float v2f __attribute__((ext_vector_type(2)));
typedef float v8f __attribute__((ext_vector_type(8)));

// ---------------------------------------------------------------- helpers
__device__ __forceinline__ float fexp_dev(float x, float e) {
    float m = powf(fabsf(x), e);
    return (x > 0.f) ? m : ((x < 0.f) ? -m : 0.f);   // sign(x)*|x|^e, sign(0)=0
}
__device__ __forceinline__ float clamp_sq(float v) {
    float sgn = (v > 0.f) ? 1.f : -1.f;              // ((v>0)*2-1)
    return sgn * fmaxf(fabsf(v), 1e-6f);
}
__device__ float block_reduce_256(float v, float* sm, int tid) {
    sm[tid] = v; __syncthreads();
    for (int off = 128; off >= 1; off >>= 1) {
        if (tid < off) sm[tid] += sm[tid + off];
        __syncthreads();
    }
    float r = sm[0];
    __syncthreads();
    return r;
}

// ------------------------------------------------- kernel 1: sample SQ points
__global__ void sq_gen_kernel(const float* __restrict__ scale,
                              const float* __restrict__ eps,
                              const float* __restrict__ etas,
                              const float* __restrict__ omegas,
                              float* __restrict__ Xp, float* __restrict__ X2p) {
    int bp = blockIdx.x;          // 0 .. B*P-1
    int s  = threadIdx.x;         // 0 .. 255
    if (s >= SPAD) return;
    float cx = 0.f, cy = 0.f, cz = 0.f, d2 = BIGD;   // pad rows: X=0, |X|^2 huge
    if (s < S_) {
        float a1 = scale[bp * 3 + 0], a2 = scale[bp * 3 + 1], a3 = scale[bp * 3 + 2];
        float e1 = eps[bp * 2 + 0],  e2 = eps[bp * 2 + 1];
        float eta = etas[bp * S_ + s];   if (eta == 0.f) eta += 1e-6f;
        float om  = omegas[bp * S_ + s]; if (om  == 0.f) om  += 1e-6f;
        float ce = cosf(eta), se = sinf(eta), co = cosf(om), so = sinf(om);
        float fe = fexp_dev(ce, e1);
        float x = a1 * fe * fexp_dev(co, e2);
        float y = a2 * fe * fexp_dev(so, e2);
        float z = a3 * fexp_dev(se, e1);
        cx = clamp_sq(x); cy = clamp_sq(y); cz = clamp_sq(z);
        d2 = cx * cx + cy * cy + cz * cz;
    }
    int o = bp * SPAD + s;
    Xp[o * 3 + 0] = cx; Xp[o * 3 + 1] = cy; Xp[o * 3 + 2] = cz;
    X2p[o] = d2;
}

// ----------------------------------- kernel 2: fused CD (WMMA) + cuboid loss
__global__ __launch_bounds__(32) void cd_cub_kernel(
        const float* __restrict__ pc,    const float* __restrict__ nm,
        const float* __restrict__ trans, const float* __restrict__ rot,
        const float* __restrict__ scale, const float* __restrict__ assign,
        const float* __restrict__ Xp,    const float* __restrict__ X2p,
        float* __restrict__ partialCD,   float* __restrict__ partialCub) {
    int wid   = blockIdx.x;             // 0 .. NWAVES-1
    int chunk = wid % NCHUNK;
    int bp    = wid / NCHUNK;
    int b = bp / P_, p = bp % P_;
    int lane = threadIdx.x, l16 = lane & 15, half = lane >> 4;

    float R[9];
    #pragma unroll
    for (int i = 0; i < 9; i++) R[i] = rot[bp * 9 + i];
    float t0 = trans[bp * 3 + 0], t1 = trans[bp * 3 + 1], t2 = trans[bp * 3 + 2];
    float s0 = scale[bp * 3 + 0], s1 = scale[bp * 3 + 1], s2 = scale[bp * 3 + 2];

    v2f   bfrag[JTILES];
    float pc2[JTILES], aw[JTILES], minD[JTILES];
    float cub_acc = 0.f;

    #pragma unroll
    for (int j = 0; j < JTILES; j++) {
        int n   = chunk * (JTILES * 16) + j * 16 + l16;
        int off = (b * N_ + n) * 3;
        // pc in primitive frame: out_i = sum_j R[j][i] * (pc_j - t_j)
        float dx = pc[off] - t0, dy = pc[off + 1] - t1, dz = pc[off + 2] - t2;
        float ix = R[0] * dx + R[3] * dy + R[6] * dz;
        float iy = R[1] * dx + R[4] * dy + R[7] * dz;
        float iz = R[2] * dx + R[5] * dy + R[8] * dz;
        pc2[j] = ix * ix + iy * iy + iz * iz;
        v2f bf;
        if (half) { bf.x = iz; bf.y = 1.0f; } else { bf.x = ix; bf.y = iy; }
        bfrag[j] = bf;
        // normals in primitive frame (reference subtracts trans here too)
        float ex = nm[off] - t0, ey = nm[off + 1] - t1, ez = nm[off + 2] - t2;
        float jx = R[0] * ex + R[3] * ey + R[6] * ez;
        float jy = R[1] * ex + R[4] * ey + R[7] * ez;
        float jz = R[2] * ex + R[5] * ey + R[8] * ez;
        // argmax over 6 face normals (+/- each axis); /norm doesn't change argmax
        float sims[6] = { -jx, jx, -jy, jy, -jz, jz };
        int f = 0; float best = sims[0];
        #pragma unroll
        for (int k = 1; k < 6; k++) { if (sims[k] > best) { best = sims[k]; f = k; } }
        int   a   = f >> 1;
        float sgn = (f & 1) ? 1.f : -1.f;
        float pi3[3] = { ix, iy, iz }, sc3[3] = { s0, s1, s2 };
        float diff = 0.f;
        #pragma unroll
        for (int i = 0; i < 3; i++) {
            float c = (i == a) ? (sgn * sc3[i] - pi3[i])
                               : fmaxf(fabsf(pi3[i]) - sc3[i], 0.f);
            diff += c * c;
        }
        float w = assign[(b * N_ + n) * P_ + p];
        aw[j] = w;
        if (half == 0) cub_acc += diff * w;
        minD[j] = BIGD;
    }

    // CD cross term via V_WMMA_F32_16X16X4_F32:
    // A[s][k] = {-2X, |X|^2}  (16x4),  B[k][n] = {pc_inver, 1} (4x16)
    // -> W[s][n] = |X|^2 - 2 X.pc ; D = W + |pc|^2
    const float* Xb  = Xp  + bp * SPAD * 3;
    const float* X2b = X2p + bp * SPAD;
    for (int t = 0; t < STILES; t++) {
        int s = t * 16 + l16;
        float x0 = Xb[s * 3 + 0], x1 = Xb[s * 3 + 1], x2 = Xb[s * 3 + 2];
        float xs = X2b[s];
        v2f af;
        if (half) { af.x = -2.f * x2; af.y = xs; }
        else      { af.x = -2.f * x0; af.y = -2.f * x1; }
        #pragma unroll
        for (int j = 0; j < JTILES; j++) {
            v8f c = {};
            v8f w = __builtin_amdgcn_wmma_f32_16x16x4_f32(
                        false, af, false, bfrag[j], (short)0, c, false, false);
            float m = w[0];
            #pragma unroll
            for (int r = 1; r < 8; r++) m = fminf(m, w[r]);
            minD[j] = fminf(minD[j], m + pc2[j]);
        }
    }

    float cd_acc = 0.f;
    #pragma unroll
    for (int j = 0; j < JTILES; j++) {
        float o = __shfl_xor(minD[j], 16, 32);       // combine M-halves
        float m = fmaxf(fminf(minD[j], o), 0.f);     // == min over s of max(D,0)
        if (half == 0) cd_acc += m * aw[j];
    }
    #pragma unroll
    for (int off = 16; off >= 1; off >>= 1) {
        cd_acc  += __shfl_xor(cd_acc,  off, 32);
        cub_acc += __shfl_xor(cub_acc, off, 32);
    }
    if (lane == 0) { partialCD[wid] = cd_acc; partialCub[wid] = cub_acc; }
}

// --------------------------------------- kernel 3: assign column sums (b,p)
__global__ void colsum_kernel(const float* __restrict__ assign,
                              float* __restrict__ colsum) {
    int bp = blockIdx.x; int b = bp / P_, p = bp % P_;
    int tid = threadIdx.x;
    float s = 0.f;
    for (int n = tid; n < N_; n += 256) s += assign[(b * N_ + n) * P_ + p];
    __shared__ float sm[256];
    float tot = block_reduce_256(s, sm, tid);
    if (tid == 0) colsum[bp] = tot;
}

// -------------------------------------------------- kernel 4: final combine
__global__ void finalize_kernel(const float* __restrict__ exist,
                                const float* __restrict__ colsum,
                                const float* __restrict__ pcd,
                                const float* __restrict__ pcub,
                                float* __restrict__ out) {
    int tid = threadIdx.x;     // 256 threads
    __shared__ float sm[256];
    float scd = 0.f, scub = 0.f;
    for (int i = tid; i < NWAVES; i += 256) { scd += pcd[i]; scub += pcub[i]; }
    float cd_sum  = block_reduce_256(scd,  sm, tid);
    float cub_sum = block_reduce_256(scub, sm, tid);

    // existence BCE, mean over B*P
    float e = 0.f;
    for (int i = tid; i < B_ * P_; i += 256) {
        float gt = (colsum[i] > EXIST_THRED_) ? 1.f : 0.f;
        float pr = exist[i];
        float l1 = fmaxf(logf(pr), -100.f);
        float l0 = fmaxf(logf(1.f - pr), -100.f);
        e += -(gt * l1 + (1.f - gt) * l0);
    }
    float ext = block_reduce_256(e, sm, tid) / (float)(B_ * P_);

    // sparsity: per b, (mean_p sqrt(colsum/N + 0.01))^2, mean over b
    float sp = 0.f;
    if (tid < B_) {
        float m = 0.f;
        for (int p = 0; p < P_; p++)
            m += sqrtf(colsum[tid * P_ + p] / (float)N_ + 0.01f);
        m /= (float)P_;
        sp = m * m;
    }
    float sps = block_reduce_256(sp, sm, tid) / (float)B_;

    if (tid == 0) {
        float inv = 1.f / (float)(B_ * N_);
        float cub = cub_sum * inv;
        float p2p = cd_sum * inv;
        float cd  = 2.f * p2p;              // p2p + prim2p (identical in ref)
        out[0] = 1.0f * cub + 1.0f * cd + 0.1f * ext + 0.1f * sps;
    }
}

// ---------------------------------------------------------------- launcher
extern "C" void kernel_launch(void* const* d_in, const int* in_sizes, int n_in,
                              void* d_out, int out_size, void* d_ws, size_t ws_size,
                              hipStream_t stream) {
    const float* pc        = (const float*)d_in[0];
    const float* normals   = (const float*)d_in[1];
    const float* trans     = (const float*)d_in[2];
    const float* rotate    = (const float*)d_in[3];
    const float* scale     = (const float*)d_in[4];
    const float* shape_eps = (const float*)d_in[5];
    const float* exist     = (const float*)d_in[6];
    const float* assign    = (const float*)d_in[7];
    const float* etas      = (const float*)d_in[8];
    const float* omegas    = (const float*)d_in[9];

    float* ws     = (float*)d_ws;
    float* Xp     = ws;                              // B*P*SPAD*3 = 79872
    float* X2p    = Xp  + B_ * P_ * SPAD * 3;        // B*P*SPAD   = 26624
    float* colsum = X2p + B_ * P_ * SPAD;            // B*P        = 128
    float* pcd    = colsum + B_ * P_;                // NWAVES     = 2048
    float* pcub   = pcd + NWAVES;                    // NWAVES     = 2048

    sq_gen_kernel<<<B_ * P_, 256, 0, stream>>>(scale, shape_eps, etas, omegas, Xp, X2p);
    cd_cub_kernel<<<NWAVES, 32, 0, stream>>>(pc, normals, trans, rotate, scale,
                                             assign, Xp, X2p, pcd, pcub);
    colsum_kernel<<<B_ * P_, 256, 0, stream>>>(assign, colsum);
    finalize_kernel<<<1, 256, 0, stream>>>(exist, colsum, pcd, pcub, (float*)d_out);
}